// DualContrastiveModel_84241488544091
// MI455X (gfx1250) — compile-verified
//
#include <hip/hip_runtime.h>
#include <cmath>

typedef float v2f __attribute__((ext_vector_type(2)));
typedef float v8f __attribute__((ext_vector_type(8)));

constexpr int kB = 32;      // batch
constexpr int kN = 200;     // session length
constexpr int kD = 100;     // embedding dim
constexpr int kHROWS = 224; // padded rows of h in LDS (>= 13*16 = 208)
constexpr int kHS = 112;    // padded row stride of h (>= 7*16)
constexpr int kAS = 112;    // padded a_k stride
constexpr int kBAND = 16;   // rows per block
constexpr int kNBANDS = 13; // 13*16 = 208 >= 200
constexpr float kNEG = -9e15f;
constexpr float kSLOPE = 0.2f;

__device__ __forceinline__ float leaky(float s) { return s > 0.f ? s : kSLOPE * s; }

__global__ __launch_bounds__(256)
void local_agg_wmma(const int* __restrict__ inputs,
                    const int* __restrict__ adj,
                    const float* __restrict__ emb,
                    const float* __restrict__ a0,
                    const float* __restrict__ a1,
                    const float* __restrict__ a2,
                    const float* __restrict__ a3,
                    float* __restrict__ out)
{
    extern __shared__ float smem[];
    float* h_s     = smem;                       // kHROWS * kHS
    float* a_s     = h_s + kHROWS * kHS;         // 4 * kAS
    float* alpha_s = a_s + 4 * kAS;              // kBAND * kN
    int*   inp_s   = (int*)(alpha_s + kBAND*kN); // kN

    const int b    = blockIdx.x;
    const int band = blockIdx.y;                 // rows [band*16, band*16+16)
    const int tid  = threadIdx.x;
    const int lane = tid & 31;
    const int wave = tid >> 5;
    const int lrow  = lane & 15;                 // M (A/D) or N (B/D) index in tile
    const int khalf = lane >> 4;                 // K offset selector

    // ---- stage 0: item ids ----
    for (int i = tid; i < kN; i += 256) inp_s[i] = inputs[b * kN + i];
    __syncthreads();

    // ---- stage 0b: gather h into LDS (zero padded), load a_k ----
    for (int t = tid; t < kHROWS * kHS; t += 256) {
        int i = t / kHS, d = t - i * kHS;
        float v = 0.f;
        if (i < kN && d < kD) v = emb[inp_s[i] * kD + d];
        h_s[t] = v;
    }
    for (int t = tid; t < 4 * kAS; t += 256) {
        int k = t / kAS, d = t - k * kAS;
        const float* ap = (k == 0) ? a0 : (k == 1) ? a1 : (k == 2) ? a2 : a3;
        a_s[t] = (d < kD) ? ap[d] : 0.f;
    }
    __syncthreads();

    // ---- phase 1: e_k = (h * a_k) @ h^T for this band, all j tiles ----
    const int i_base = band * kBAND;
    for (int jt = wave; jt < 13; jt += 8) {
        v8f acc0 = {}, acc1 = {}, acc2 = {}, acc3 = {};
        const int arow = i_base + lrow;   // A fragment row (band rows)
        const int brow = jt * 16 + lrow;  // B fragment row (neighbor rows)
        for (int kk = 0; kk < 25; ++kk) {
            const int d0 = kk * 4 + 2 * khalf;
            v2f hv, bv;
            hv[0] = h_s[arow * kHS + d0];
            hv[1] = h_s[arow * kHS + d0 + 1];
            bv[0] = h_s[brow * kHS + d0];
            bv[1] = h_s[brow * kHS + d0 + 1];
            v2f s0, s1, s2, s3;
            s0[0] = hv[0] * a_s[0*kAS + d0]; s0[1] = hv[1] * a_s[0*kAS + d0 + 1];
            s1[0] = hv[0] * a_s[1*kAS + d0]; s1[1] = hv[1] * a_s[1*kAS + d0 + 1];
            s2[0] = hv[0] * a_s[2*kAS + d0]; s2[1] = hv[1] * a_s[2*kAS + d0 + 1];
            s3[0] = hv[0] * a_s[3*kAS + d0]; s3[1] = hv[1] * a_s[3*kAS + d0 + 1];
            acc0 = __builtin_amdgcn_wmma_f32_16x16x4_f32(false, s0, false, bv, (short)0, acc0, false, false);
            acc1 = __builtin_amdgcn_wmma_f32_16x16x4_f32(false, s1, false, bv, (short)0, acc1, false, false);
            acc2 = __builtin_amdgcn_wmma_f32_16x16x4_f32(false, s2, false, bv, (short)0, acc2, false, false);
            acc3 = __builtin_amdgcn_wmma_f32_16x16x4_f32(false, s3, false, bv, (short)0, acc3, false, false);
        }
        // leaky-relu + adjacency select -> logits in LDS
        #pragma unroll
        for (int e = 0; e < 8; ++e) {
            int m = e + 8 * khalf;     // local band row
            int i = i_base + m;
            int j = jt * 16 + lrow;
            if (i < kN && j < kN) {
                int av = adj[(b * kN + i) * kN + j];
                float val = kNEG;
                if      (av == 1) val = leaky(acc0[e]);
                else if (av == 2) val = leaky(acc1[e]);
                else if (av == 3) val = leaky(acc2[e]);
                else if (av == 4) val = leaky(acc3[e]);
                alpha_s[m * kN + j] = val;
            }
        }
    }
    __syncthreads();

    // ---- phase 1.5: row softmax over j (wave32 shuffle reductions) ----
    for (int r = wave; r < kBAND; r += 8) {
        float mx = -3.4e38f;
        for (int j = lane; j < kN; j += 32) mx = fmaxf(mx, alpha_s[r * kN + j]);
        #pragma unroll
        for (int off = 16; off > 0; off >>= 1) mx = fmaxf(mx, __shfl_xor(mx, off, 32));
        float sum = 0.f;
        for (int j = lane; j < kN; j += 32) {
            float ex = expf(alpha_s[r * kN + j] - mx);
            alpha_s[r * kN + j] = ex;
            sum += ex;
        }
        #pragma unroll
        for (int off = 16; off > 0; off >>= 1) sum += __shfl_xor(sum, off, 32);
        float rs = 1.f / sum;
        for (int j = lane; j < kN; j += 32) alpha_s[r * kN + j] *= rs;
    }
    __syncthreads();

    // ---- phase 2: out[band rows] = alpha @ h  (M=16, K=200, N=100->112) ----
    for (int dt = wave; dt < 7; dt += 8) {
        v8f acc = {};
        for (int kk = 0; kk < 50; ++kk) {
            const int j0 = kk * 4 + 2 * khalf;
            v2f av, bv;
            av[0] = alpha_s[lrow * kN + j0];
            av[1] = alpha_s[lrow * kN + j0 + 1];
            bv[0] = h_s[j0 * kHS + dt * 16 + lrow];
            bv[1] = h_s[(j0 + 1) * kHS + dt * 16 + lrow];
            acc = __builtin_amdgcn_wmma_f32_16x16x4_f32(false, av, false, bv, (short)0, acc, false, false);
        }
        #pragma unroll
        for (int e = 0; e < 8; ++e) {
            int m = e + 8 * khalf;
            int i = i_base + m;
            int d = dt * 16 + lrow;
            if (i < kN && d < kD) out[(b * kN + i) * kD + d] = acc[e];
        }
    }
}

extern "C" void kernel_launch(void* const* d_in, const int* in_sizes, int n_in,
                              void* d_out, int out_size, void* d_ws, size_t ws_size,
                              hipStream_t stream) {
    const int*   inputs = (const int*)  d_in[0];
    const int*   adj    = (const int*)  d_in[1];
    // d_in[2] = mask_item (unused by the reference computation in eval mode)
    const float* emb    = (const float*)d_in[3];
    const float* a0     = (const float*)d_in[4];
    const float* a1     = (const float*)d_in[5];
    const float* a2     = (const float*)d_in[6];
    const float* a3     = (const float*)d_in[7];
    float* out = (float*)d_out;

    const size_t shmem = (size_t)(kHROWS * kHS + 4 * kAS + kBAND * kN) * sizeof(float)
                       + (size_t)kN * sizeof(int);   // ~115.7 KB -> 2 blocks/WGP
    dim3 grid(kB, kNBANDS);
    local_agg_wmma<<<grid, 256, shmem, stream>>>(inputs, adj, emb, a0, a1, a2, a3, out);
}